// VQAModule_40965398069252
// MI455X (gfx1250) — compile-verified
//
#include <hip/hip_runtime.h>
#include <math.h>

#define B_N 4096
#define S_N 200
#define D_N 128
#define C_N 64
#define NEG_INF (-1e30f)
__constant__ constexpr float SCALE_INV = 0.08838834764831845f; // 1/sqrt(128)

typedef __attribute__((ext_vector_type(2))) float v2f;
typedef __attribute__((ext_vector_type(8))) float v8f;
typedef __attribute__((ext_vector_type(4))) int v4i;

// ---------------------------------------------------------------------------
// WMMA f32 16x16x4 wrapper (CDNA5: D = A(16x4) * B(4x16) + C, fp32 throughout)
// ---------------------------------------------------------------------------
__device__ __forceinline__ v8f wmma_f32_16x16x4(v2f a, v2f b, v8f c) {
#if defined(__HIP_DEVICE_COMPILE__)
#if __has_builtin(__builtin_amdgcn_wmma_f32_16x16x4_f32)
  return __builtin_amdgcn_wmma_f32_16x16x4_f32(false, a, false, b, (short)0, c,
                                               false, false);
#else
#error "missing __builtin_amdgcn_wmma_f32_16x16x4_f32 on device toolchain"
#endif
#else
  return c;  // host pass: parsed but never executed
#endif
}

// One wave computes a 16xN tile of D = A(16xK) * B, scaled by alpha.
// A: row-major [16 x K], lda.
// B: if BT, stored as B[n][k] (ldb strides n)  -> D = A * B^T  (torch Linear)
//    else    stored as B[k][n] (ldb strides k) -> D = A * B
// Fragment layouts per CDNA5 ISA 7.12.2 (wave32):
//   A frag: lane l (l<16): a[i] = A[l][2*half + i + k0]
//   B frag: lane l (l<16): b[i] = B[2*half + i + k0][l]
//   D frag: vgpr j -> row (j + 8*half), col = lane&15.
template <bool BT>
__device__ __forceinline__ void gemm16(const float* __restrict__ A, int lda,
                                       const float* __restrict__ Bm, int ldb,
                                       float* __restrict__ Dst, int ldd,
                                       int K, int N, int lane, float alpha) {
  const int hw = lane >> 4;   // 0/1: which K-pair / which row-half of D
  const int lr = lane & 15;   // row of A frag / col of B,D frag
  for (int n0 = 0; n0 < N; n0 += 16) {
    v8f acc = {};
    for (int k0 = 0; k0 < K; k0 += 4) {
      const int ka = k0 + 2 * hw;
      v2f a;
      a.x = A[lr * lda + ka];
      a.y = A[lr * lda + ka + 1];
      v2f b;
      if (BT) {
        b.x = Bm[(n0 + lr) * ldb + ka];
        b.y = Bm[(n0 + lr) * ldb + ka + 1];
      } else {
        b.x = Bm[ka * ldb + n0 + lr];
        b.y = Bm[(ka + 1) * ldb + n0 + lr];
      }
      acc = wmma_f32_16x16x4(a, b, acc);
    }
#pragma unroll
    for (int j = 0; j < 8; ++j)
      Dst[(j + 8 * hw) * ldd + n0 + lr] = acc[j] * alpha;
  }
}

// ---------------------------------------------------------------------------
// Kernel 1: masked mean-pool.  One wave per basket row, float4 streaming.
// ---------------------------------------------------------------------------
__global__ void pool_kernel(const float* __restrict__ items,
                            const unsigned char* __restrict__ mask,
                            float* __restrict__ basket) {
  const int wave = threadIdx.x >> 5;
  const int lane = threadIdx.x & 31;
  const int b = blockIdx.x * 8 + wave;
  const float4* src = (const float4*)(items + (size_t)b * S_N * D_N);
  const unsigned char* m = mask + (size_t)b * S_N;
  float4 acc = {0.f, 0.f, 0.f, 0.f};
  float cnt = 0.f;
#pragma unroll 4
  for (int s = 0; s < S_N; ++s) {
    float4 v = src[s * (D_N / 4) + lane];
    float f = m[s] ? 1.0f : 0.0f;
    acc.x = fmaf(v.x, f, acc.x);
    acc.y = fmaf(v.y, f, acc.y);
    acc.z = fmaf(v.z, f, acc.z);
    acc.w = fmaf(v.w, f, acc.w);
    cnt += f;
  }
  const float inv = 1.0f / fmaxf(cnt, 1.0f);
  float4 o = {acc.x * inv, acc.y * inv, acc.z * inv, acc.w * inv};
  ((float4*)(basket + (size_t)b * D_N))[lane] = o;
}

// ---------------------------------------------------------------------------
// Kernel 2a: k1 = codebook @ W_k1^T   [64 x 128], 4 waves, 1 block.
// ---------------------------------------------------------------------------
__global__ void k1_kernel(const float* __restrict__ codebook,
                          const float* __restrict__ W_k1,
                          float* __restrict__ k1) {
  const int wave = threadIdx.x >> 5;
  const int lane = threadIdx.x & 31;
  gemm16<true>(codebook + wave * 16 * D_N, D_N, W_k1, D_N,
               k1 + wave * 16 * D_N, D_N, D_N, D_N, lane, 1.0f);
}

// ---------------------------------------------------------------------------
// Kernel 2: fused GEMM chain per 16-row tile (one wave per tile):
//   q1 = basket@Wq1^T; attn1 = softmax(q1@k1^T/s); pattern = attn1@codebook;
//   q2 = pattern@Wq2^T; qk2 = (q2@Wk2)/s
// ---------------------------------------------------------------------------
__global__ void chain_kernel(const float* __restrict__ basket,
                             const float* __restrict__ k1,
                             const float* __restrict__ codebook,
                             const float* __restrict__ W_q1,
                             const float* __restrict__ W_q2,
                             const float* __restrict__ W_k2,
                             float* __restrict__ qk2) {
  __shared__ float bufA[8][16 * D_N];  // 64 KB
  __shared__ float bufB[8][16 * D_N];  // 64 KB
  const int wave = threadIdx.x >> 5;
  const int lane = threadIdx.x & 31;
  const int tile = blockIdx.x * 8 + wave;
  float* a = bufA[wave];
  float* bb = bufB[wave];

  // q1 -> a [16x128]
  gemm16<true>(basket + (size_t)tile * 16 * D_N, D_N, W_q1, D_N, a, D_N,
               D_N, D_N, lane, 1.0f);
  __syncthreads();
  // logits1 = q1 @ k1^T / sqrt(D) -> bb [16x64]
  gemm16<true>(a, D_N, k1, D_N, bb, C_N, D_N, C_N, lane, SCALE_INV);
  __syncthreads();
  // row softmax over 64 cols: 2 lanes per row, half-wave shuffle combine
  {
    const int row = lane & 15;
    const int part = lane >> 4;
    float* r = bb + row * C_N + part * 32;
    float mx = NEG_INF;
    for (int i = 0; i < 32; ++i) mx = fmaxf(mx, r[i]);
    mx = fmaxf(mx, __shfl_xor(mx, 16));
    float sum = 0.f;
    for (int i = 0; i < 32; ++i) {
      float e = __expf(r[i] - mx);
      r[i] = e;
      sum += e;
    }
    sum += __shfl_xor(sum, 16);
    const float inv = 1.0f / sum;
    for (int i = 0; i < 32; ++i) r[i] *= inv;
  }
  __syncthreads();
  // pattern = attn1 @ codebook -> a [16x128]  (B row-major, K=64)
  gemm16<false>(bb, C_N, codebook, D_N, a, D_N, C_N, D_N, lane, 1.0f);
  __syncthreads();
  // q2 = pattern @ W_q2^T -> bb [16x128]
  gemm16<true>(a, D_N, W_q2, D_N, bb, D_N, D_N, D_N, lane, 1.0f);
  __syncthreads();
  // qk2 = q2 @ W_k2 / sqrt(D) -> global  (B row-major)
  gemm16<false>(bb, D_N, W_k2, D_N, qk2 + (size_t)tile * 16 * D_N, D_N,
                D_N, D_N, lane, SCALE_INV);
}

// ---------------------------------------------------------------------------
// Kernel 3: per-row attention over items, single HBM pass via 102.4 KB LDS
// tile (fits WGP's 320 KB).  ctx0[b] = softmax(qk2[b]·item[b,s]) · item[b]
// ---------------------------------------------------------------------------
__global__ void attn2_kernel(const float* __restrict__ items,
                             const unsigned char* __restrict__ mask,
                             const float* __restrict__ qk2,
                             float* __restrict__ ctx0) {
  __shared__ __align__(16) float tile[S_N * D_N];  // 102400 B
  __shared__ float q[D_N];
  __shared__ float logits[S_N];
  __shared__ float red[8];
  __shared__ float part[D_N];
  const int b = blockIdx.x;
  const int t = threadIdx.x;  // 256 threads = 8 waves
  const float* src = items + (size_t)b * S_N * D_N;

  // Stage item tile into LDS (async DMA path on gfx1250).
#if __has_builtin(__builtin_amdgcn_global_load_async_to_lds_b128)
  {
    typedef __attribute__((address_space(1))) v4i gv4i;
    typedef __attribute__((address_space(3))) v4i lv4i;
    gv4i* gp = (gv4i*)(void*)src;
    lv4i* lp = (lv4i*)(void*)tile;
    for (int i = t; i < (S_N * D_N) / 4; i += 256)
      __builtin_amdgcn_global_load_async_to_lds_b128(gp + i, lp + i, 0, 0);
  }
#if __has_builtin(__builtin_amdgcn_s_wait_asynccnt)
  __builtin_amdgcn_s_wait_asynccnt(0);
#else
  asm volatile("s_wait_asynccnt 0" ::: "memory");
#endif
#else
  for (int i = t; i < (S_N * D_N) / 4; i += 256)
    ((float4*)tile)[i] = ((const float4*)src)[i];
#endif
  if (t < D_N) q[t] = qk2[(size_t)b * D_N + t];
  __syncthreads();

  // logits[s] = q · item[s]   (qk2 already carries 1/sqrt(D))
  for (int s = t; s < S_N; s += 256) {
    float dot = 0.f;
    for (int d = 0; d < D_N; ++d) dot = fmaf(q[d], tile[s * D_N + d], dot);
    logits[s] = mask[(size_t)b * S_N + s] ? dot : NEG_INF;
  }
  __syncthreads();

  // block softmax over 200 entries
  float v = (t < S_N) ? logits[t] : NEG_INF;
  for (int o = 16; o > 0; o >>= 1) v = fmaxf(v, __shfl_xor(v, o));
  if ((t & 31) == 0) red[t >> 5] = v;
  __syncthreads();
  float mx = red[0];
  for (int i = 1; i < 8; ++i) mx = fmaxf(mx, red[i]);
  float e = (t < S_N) ? __expf(logits[t] - mx) : 0.f;
  float sv = e;
  for (int o = 16; o > 0; o >>= 1) sv += __shfl_xor(sv, o);
  __syncthreads();
  if ((t & 31) == 0) red[t >> 5] = sv;
  __syncthreads();
  float sum = 0.f;
  for (int i = 0; i < 8; ++i) sum += red[i];
  const float inv = 1.0f / sum;
  if (t < S_N) logits[t] = e * inv;
  __syncthreads();

  // ctx0[b,d] = sum_s w[s] * item[s,d]; 2 threads per d, combine via LDS
  const int d = t & (D_N - 1);
  const int h = t >> 7;
  float acc = 0.f;
  for (int s = h; s < S_N; s += 2) acc = fmaf(logits[s], tile[s * D_N + d], acc);
  if (h == 1) part[d] = acc;
  __syncthreads();
  if (h == 0) ctx0[(size_t)b * D_N + d] = acc + part[d];
}

// ---------------------------------------------------------------------------
// Kernel 4: context = ctx0 @ W_v2^T   [4096 x 128], WMMA, one wave per tile
// ---------------------------------------------------------------------------
__global__ void out_kernel(const float* __restrict__ ctx0,
                           const float* __restrict__ W_v2,
                           float* __restrict__ out) {
  const int wave = threadIdx.x >> 5;
  const int lane = threadIdx.x & 31;
  const int tile = blockIdx.x * 8 + wave;
  gemm16<true>(ctx0 + (size_t)tile * 16 * D_N, D_N, W_v2, D_N,
               out + (size_t)tile * 16 * D_N, D_N, D_N, D_N, lane, 1.0f);
}

// ---------------------------------------------------------------------------
extern "C" void kernel_launch(void* const* d_in, const int* in_sizes, int n_in,
                              void* d_out, int out_size, void* d_ws,
                              size_t ws_size, hipStream_t stream) {
  const float* items = (const float*)d_in[0];
  const unsigned char* mask = (const unsigned char*)d_in[1];
  const float* codebook = (const float*)d_in[2];
  const float* W_q1 = (const float*)d_in[3];
  const float* W_k1 = (const float*)d_in[4];
  const float* W_q2 = (const float*)d_in[5];
  const float* W_k2 = (const float*)d_in[6];
  const float* W_v2 = (const float*)d_in[7];
  float* out = (float*)d_out;

  float* basket = (float*)d_ws;                 // B*D
  float* k1 = basket + (size_t)B_N * D_N;       // C*D
  float* qk2 = k1 + (size_t)C_N * D_N;          // B*D
  float* ctx0 = qk2 + (size_t)B_N * D_N;        // B*D

  pool_kernel<<<B_N / 8, 256, 0, stream>>>(items, mask, basket);
  k1_kernel<<<1, 128, 0, stream>>>(codebook, W_k1, k1);
  chain_kernel<<<B_N / 128, 256, 0, stream>>>(basket, k1, codebook, W_q1, W_q2,
                                              W_k2, qk2);
  attn2_kernel<<<B_N, 256, 0, stream>>>(items, mask, qk2, ctx0);
  out_kernel<<<B_N / 128, 256, 0, stream>>>(ctx0, W_v2, out);
}